// ManualLinearRegression_85693187490128
// MI455X (gfx1250) — compile-verified
//
#include <hip/hip_runtime.h>
#include <cstddef>
#include <cstdint>

#define RDIM 128
#define LDP  130   // padded LDS row stride (floats) for the WMMA kernel

typedef __attribute__((ext_vector_type(2))) float v2f;
typedef __attribute__((ext_vector_type(8))) float v8f;

// ---------------------------------------------------------------------------
// Kernel 1: per-frequency complex Gauss-Jordan inversion of (i*w*I - A).
// One workgroup per frequency; 128x128 complex matrix lives in LDS (128KB).
// Writes X = inv(i*w*I - A) as split re/im planes into d_ws.
// ---------------------------------------------------------------------------
__global__ __launch_bounds__(256) void csd_invert_kernel(
    const float* __restrict__ freqs, const float* __restrict__ A,
    float* __restrict__ ws) {
  extern __shared__ char smem_raw[];
  float2* M    = reinterpret_cast<float2*>(smem_raw);  // RDIM*RDIM complexes
  float2* rowk = M + RDIM * RDIM;                      // scaled pivot row
  float2* colk = rowk + RDIM;                          // pivot column (f_i)

  const int f   = blockIdx.x;
  const int tid = threadIdx.x;
  const float w = freqs[f];

  // M = i*w*I - A
  for (int idx = tid; idx < RDIM * RDIM; idx += 256) {
    const int r = idx >> 7;
    const int c = idx & (RDIM - 1);
    float2 v;
    v.x = -A[idx];
    v.y = (r == c) ? w : 0.0f;
    M[idx] = v;
  }
  __syncthreads();

  for (int k = 0; k < RDIM; ++k) {
    // every thread computes pivot reciprocal (LDS broadcast read)
    const float2 p = M[k * RDIM + k];
    const float inv = 1.0f / (p.x * p.x + p.y * p.y);
    float2 pinv;
    pinv.x = p.x * inv;
    pinv.y = -p.y * inv;
    __syncthreads();  // pivot read done before row/col get zeroed

    if (tid < RDIM) {
      const int t = tid;
      float2 rk, ck;
      if (t == k) {
        rk = pinv;               // rowk[k] = 1/p
        ck.x = -1.0f; ck.y = 0.0f; // colk[k] = -1 -> row k regenerates itself
      } else {
        const float2 m = M[k * RDIM + t];
        rk.x = pinv.x * m.x - pinv.y * m.y;   // scaled pivot row
        rk.y = pinv.x * m.y + pinv.y * m.x;
        ck = M[t * RDIM + k];                 // f_i
      }
      rowk[t] = rk;
      colk[t] = ck;
      const float2 z = {0.0f, 0.0f};
      M[k * RDIM + t] = z;   // zero pivot row & column so the update is uniform
      M[t * RDIM + k] = z;
    }
    __syncthreads();

    // uniform rank-1 update: M[i][j] -= colk[i] * rowk[j] over all 128x128.
    // thread t -> column (t&127), rows [ (t>>7)*64 , +64 ): conflict-free LDS.
    const int j  = tid & (RDIM - 1);
    const int i0 = (tid >> 7) * 64;
    const float2 rj = rowk[j];
#pragma unroll 8
    for (int ii = 0; ii < 64; ++ii) {
      const int i = i0 + ii;
      const float2 ci = colk[i];
      float2 m = M[i * RDIM + j];
      m.x -= ci.x * rj.x - ci.y * rj.y;
      m.y -= ci.x * rj.y + ci.y * rj.x;
      M[i * RDIM + j] = m;
    }
    __syncthreads();
  }

  // write X as split planes: ws[f] = { Xr[128*128], Xi[128*128] }
  float* wsXr = ws + (size_t)f * (2 * RDIM * RDIM);
  float* wsXi = wsXr + RDIM * RDIM;
  for (int idx = tid; idx < RDIM * RDIM; idx += 256) {
    const float2 v = M[idx];
    wsXr[idx] = v.x;
    wsXi[idx] = v.y;
  }
}

// ---------------------------------------------------------------------------
// Kernel 2: res = |hrf|^2 * X diag(g_v) X^H + diag(g_e), via f32 WMMA.
// Uses X_t = X^H (A real => inv(-iwI - A^T) = inv(iwI - A)^H).
// One workgroup per frequency; X staged in LDS; 8 waves x 8 tiles of 16x16.
// Complex GEMM = 4 real V_WMMA_F32_16X16X4_F32 accumulations per K-step.
// ---------------------------------------------------------------------------
__global__ __launch_bounds__(256) void csd_wmma_kernel(
    const float* __restrict__ freqs,
    const float* __restrict__ alphas_e, const float* __restrict__ betas_e,
    const float* __restrict__ alphas_v, const float* __restrict__ betas_v,
    const float* __restrict__ ws, float2* __restrict__ out, int F) {
  extern __shared__ char smem_raw[];
  float* Xr  = reinterpret_cast<float*>(smem_raw);  // RDIM x LDP (padded)
  float* Xi  = Xr + RDIM * LDP;
  float* gsv = Xi + RDIM * LDP;  // |hrf|^2 * g_v  (RDIM floats)
  float* ge  = gsv + RDIM;       // g_e            (RDIM floats)

  const int f   = blockIdx.x;
  const int tid = threadIdx.x;
  const float w = freqs[f];

  // |hrf|^2 = |6(1+iw)^10 - 1|^2 / (25 |1+iw|^32), via polar form.
  const float L    = 1.0f + w * w;          // |1+iw|^2
  const float L5   = (L * L) * (L * L) * L; // L^5 = |1+iw|^10
  const float L16  = L5 * L5 * L5 * L;      // L^16 = |1+iw|^32
  const float ang  = 10.0f * atanf(w);
  const float re10 = L5 * cosf(ang);
  const float im10 = L5 * sinf(ang);
  const float nr   = 6.0f * re10 - 1.0f;
  const float ni   = 6.0f * im10;
  const float s    = (nr * nr + ni * ni) / (25.0f * L16);

  if (tid < RDIM) {
    const float lw = logf(w);
    gsv[tid] = s * alphas_v[tid] * expf(-betas_v[tid] * lw);  // fold |hrf|^2 in
    ge[tid]  = alphas_e[tid] * expf(-betas_e[tid] * lw);
  }

  // stage X planes into padded LDS
  const float* wr = ws + (size_t)f * (2 * RDIM * RDIM);
  const float* wi = wr + RDIM * RDIM;
  for (int idx = tid; idx < RDIM * RDIM; idx += 256) {
    const int r = idx >> 7;
    const int c = idx & (RDIM - 1);
    Xr[r * LDP + c] = wr[idx];
    Xi[r * LDP + c] = wi[idx];
  }
  __syncthreads();

  const int lane = tid & 31;
  const int lr   = lane & 15;
  const int kh   = (lane >> 4) << 1;  // 0 or 2: lane-half K offset (f32 16x4 layout)
  const int m0   = (tid >> 5) * 16;   // wave -> tile row strip

  const float* Ar = Xr + (m0 + lr) * LDP;
  const float* Ai = Xi + (m0 + lr) * LDP;

  for (int nt = 0; nt < 8; ++nt) {
    const int n0 = nt * 16;
    const float* Br = Xr + (n0 + lr) * LDP;  // B-frag of X^T == A-frag of X
    const float* Bi = Xi + (n0 + lr) * LDP;
    v8f accR = {};
    v8f accI = {};
#pragma unroll 4
    for (int k0 = 0; k0 < RDIM; k0 += 4) {
      const int k = k0 + kh;
      const float g0 = gsv[k];
      const float g1 = gsv[k + 1];
      const v2f xa = *reinterpret_cast<const v2f*>(Ar + k);  // Xr[m][k..k+1]
      const v2f ya = *reinterpret_cast<const v2f*>(Ai + k);  // Xi[m][k..k+1]
      const v2f xb = *reinterpret_cast<const v2f*>(Br + k);  // Xr[n][k..k+1]
      const v2f yb = *reinterpret_cast<const v2f*>(Bi + k);  // Xi[n][k..k+1]
      v2f ar;  ar.x  = xa.x * g0;  ar.y  = xa.y * g1;   // Wr = Xr * g
      v2f ai;  ai.x  = ya.x * g0;  ai.y  = ya.y * g1;   // Wi = Xi * g
      v2f nar; nar.x = -ar.x;      nar.y = -ar.y;
      // B = W X^H :  Br_acc = Wr*Xr^T + Wi*Xi^T ; Bi_acc = Wi*Xr^T - Wr*Xi^T
      accR = __builtin_amdgcn_wmma_f32_16x16x4_f32(false, ar,  false, xb, (short)0, accR, false, false);
      accR = __builtin_amdgcn_wmma_f32_16x16x4_f32(false, ai,  false, yb, (short)0, accR, false, false);
      accI = __builtin_amdgcn_wmma_f32_16x16x4_f32(false, ai,  false, xb, (short)0, accI, false, false);
      accI = __builtin_amdgcn_wmma_f32_16x16x4_f32(false, nar, false, yb, (short)0, accI, false, false);
    }

    // epilogue: + diag(g_e), store transposed (R^2, F) complex64 layout
    const int nn    = n0 + lr;
    const int mbase = m0 + ((lane >> 4) << 3);  // C layout: lanes>=16 -> M+8
#pragma unroll
    for (int v = 0; v < 8; ++v) {
      const int mm = mbase + v;
      float re = accR[v];
      if (mm == nn) re += ge[mm];
      const size_t idx = (size_t)(mm * RDIM + nn) * (size_t)F + (size_t)f;
      float2 o;
      o.x = re;
      o.y = accI[v];
      out[idx] = o;
    }
  }
}

// ---------------------------------------------------------------------------
extern "C" void kernel_launch(void* const* d_in, const int* in_sizes, int n_in,
                              void* d_out, int out_size, void* d_ws, size_t ws_size,
                              hipStream_t stream) {
  const float* freqs    = (const float*)d_in[0];
  const float* alphas_e = (const float*)d_in[1];
  const float* betas_e  = (const float*)d_in[2];
  const float* alphas_v = (const float*)d_in[3];
  const float* betas_v  = (const float*)d_in[4];
  const float* A        = (const float*)d_in[5];
  const int F = in_sizes[0];  // 2048

  float*  ws  = (float*)d_ws;   // needs F * 2 * R*R * 4B = 256 MB
  float2* out = (float2*)d_out; // complex64 as interleaved float2

  const size_t lds1 = (size_t)(RDIM * RDIM + 2 * RDIM) * sizeof(float2); // ~130 KB
  const size_t lds2 = (size_t)(2 * RDIM * LDP + 2 * RDIM) * sizeof(float); // ~131 KB
  (void)hipFuncSetAttribute(reinterpret_cast<const void*>(csd_invert_kernel),
                            hipFuncAttributeMaxDynamicSharedMemorySize, (int)lds1);
  (void)hipFuncSetAttribute(reinterpret_cast<const void*>(csd_wmma_kernel),
                            hipFuncAttributeMaxDynamicSharedMemorySize, (int)lds2);

  csd_invert_kernel<<<F, 256, lds1, stream>>>(freqs, A, ws);
  csd_wmma_kernel<<<F, 256, lds2, stream>>>(freqs, alphas_e, betas_e,
                                            alphas_v, betas_v, ws, out, F);
}